// MultiHeadAttention_54743653154933
// MI455X (gfx1250) — compile-verified
//
#include <hip/hip_runtime.h>
#include <hip/hip_bf16.h>
#include <math.h>

typedef float v2f __attribute__((ext_vector_type(2)));
typedef float v8f __attribute__((ext_vector_type(8)));

#define HID   4096
#define NH    32
#define NKV   8
#define HD    128
#define BS    32
#define BPS   32     // blocks per seq
#define BLK   128    // block size (tokens)
#define QKVN  6144   // 4096 + 1024 + 1024
#define WAVES 8

// ---------------------------------------------------------------------------
// 32xN GEMM tile: one wave computes a 32x16 slab of C = A(32xK) * B(KxN)
// using V_WMMA_F32_16X16X4_F32. Two accumulators share one B fragment so
// every weight element is loaded exactly once. LDA/LDB are compile-time so
// the unrolled loop body is pure immediate-offset loads + pointer bumps.
// A frag (16x4 f32): lanes 0-15 rows M, VGPR0=K0/VGPR1=K1; lanes 16-31 K2/K3.
// B frag (4x16 f32): lanes 0-15 cols N, VGPR0=K0/VGPR1=K1; lanes 16-31 K2/K3.
// C frag: VGPR j -> M=j (lanes 0-15) / M=j+8 (lanes 16-31), N = lane&15.
// ---------------------------------------------------------------------------
template<int LDA, int LDB>
__device__ __forceinline__ void gemm_tile32(const float* __restrict__ A,
                                            const float* __restrict__ B, int bn,
                                            float* __restrict__ C, int ldc, int n0, int K)
{
  const int lane = threadIdx.x & 31;
  const int l16  = lane & 15;
  const int koff = (lane >> 4) * 2;   // lanes 16-31 handle K+2 / K+3

  v8f c0 = {0,0,0,0,0,0,0,0};
  v8f c1 = {0,0,0,0,0,0,0,0};

  const float* a0p = A + (size_t)l16        * LDA + koff;
  const float* a1p = A + (size_t)(l16 + 16) * LDA + koff;
  const float* bp  = B + (size_t)koff * LDB + bn + l16;

#pragma unroll 8
  for (int k = 0; k < K; k += 4) {
    v2f a0 = { a0p[0], a0p[1] };
    v2f a1 = { a1p[0], a1p[1] };
    v2f bb = { bp[0], bp[LDB] };
    // prefetch the weight stream 128 rows (8 KB) ahead
    __builtin_prefetch((const void*)(bp + (size_t)128 * LDB), 0, 0);
    c0 = __builtin_amdgcn_wmma_f32_16x16x4_f32(false, a0, false, bb, (short)0, c0, false, false);
    c1 = __builtin_amdgcn_wmma_f32_16x16x4_f32(false, a1, false, bb, (short)0, c1, false, false);
    a0p += 4;
    a1p += 4;
    bp  += (size_t)4 * LDB;
  }

  const int rbase = (lane >> 4) * 8;
#pragma unroll
  for (int j = 0; j < 8; ++j) {
    C[(size_t)(rbase + j)      * ldc + n0 + l16] = c0[j];
    C[(size_t)(rbase + j + 16) * ldc + n0 + l16] = c1[j];
  }
}

// ---------------------------------------------------------------------------
// Kernel 1: QKV projection. hidden(32x4096) @ [Wq|Wk|Wv] -> qkv ws (32x6144)
// grid 48 blocks x 8 waves; each wave owns one 16-col tile.
// ---------------------------------------------------------------------------
__global__ __launch_bounds__(256) void qkv_gemm_kernel(const float* __restrict__ A,
                                                       const float* __restrict__ Wq,
                                                       const float* __restrict__ Wk,
                                                       const float* __restrict__ Wv,
                                                       float* __restrict__ C)
{
  const int tile = blockIdx.x * WAVES + (threadIdx.x >> 5);
  const int n0 = tile * 16;
  if (n0 < 4096) {
    gemm_tile32<HID, 4096>(A, Wq, n0,        C, QKVN, n0, HID);
  } else if (n0 < 5120) {
    gemm_tile32<HID, 1024>(A, Wk, n0 - 4096, C, QKVN, n0, HID);
  } else {
    gemm_tile32<HID, 1024>(A, Wv, n0 - 5120, C, QKVN, n0, HID);
  }
}

// ---------------------------------------------------------------------------
// Kernel 2: RMS-norm + RoPE applied in place to q (32 heads) and k (8 heads)
// inside the qkv workspace. One 128-thread block per (batch, head).
// ---------------------------------------------------------------------------
__global__ __launch_bounds__(128) void normrope_kernel(float* __restrict__ qkv,
                                                       const float* __restrict__ cosd,
                                                       const float* __restrict__ sind,
                                                       const float* __restrict__ qw,
                                                       const float* __restrict__ kw)
{
  __shared__ float red[128];
  const int h = blockIdx.x;   // 0..39 : 32 q heads then 8 k heads
  const int b = blockIdx.y;
  const int t = threadIdx.x;

  float* vec; const float* w;
  if (h < NH) { vec = qkv + (size_t)b * QKVN + (size_t)h * HD;               w = qw; }
  else        { vec = qkv + (size_t)b * QKVN + 4096 + (size_t)(h - NH) * HD; w = kw; }

  const float x = vec[t];
  red[t] = x * x;
  __syncthreads();
  for (int s = 64; s > 0; s >>= 1) {
    if (t < s) red[t] += red[t + s];
    __syncthreads();
  }
  const float rms = rsqrtf(red[0] * (1.0f / 128.0f) + 1e-6f);

  const int p   = t ^ 64;
  const float xp = vec[p];                    // read partner before any write
  const float sgn = (t < 64) ? -1.0f : 1.0f;
  const float y  = x  * rms * w[t];
  const float yp = xp * rms * w[p];
  const float o  = y * cosd[b * HD + t] + sgn * yp * sind[b * HD + t];
  __syncthreads();                            // all partner reads done
  vec[t] = o;
}

// ---------------------------------------------------------------------------
// Kernel 3: flash-decode attention, one workgroup per (kv head, batch).
// 8 waves each own strided 16-token tiles with private online softmax;
// partials merged across waves via LDS. Token pos=seqlen-1 is substituted
// with the freshly projected k/v (reference's cache .set without mutating
// the input caches).
// ---------------------------------------------------------------------------
__global__ __launch_bounds__(256) void attn_kernel(const float* __restrict__ qkv,
                                                   const float* __restrict__ key_cache,
                                                   const float* __restrict__ value_cache,
                                                   const int*   __restrict__ seq_lens,
                                                   const int*   __restrict__ block_table,
                                                   float* __restrict__ attn_out)
{
  __shared__ float ptile[WAVES * 256];      // per-wave 16x16 P tile
  __shared__ float accb [WAVES * 4 * 128];  // per-wave 4x128 partial output
  __shared__ float mb   [WAVES * 4];
  __shared__ float lb   [WAVES * 4];

  const int kvh  = blockIdx.x;              // 8
  const int b    = blockIdx.y;              // 32
  const int tid  = threadIdx.x;
  const int wave = tid >> 5;
  const int lane = tid & 31;
  const int half = lane >> 4;
  const int l16  = lane & 15;
  const int koff2 = half * 2;

  const int   seqlen = seq_lens[b];
  const int   pos    = seqlen - 1;
  const float scale  = 0.08838834764831845f;   // 1/sqrt(128)

  const float* qbase = qkv + (size_t)b * QKVN + (size_t)kvh * 4 * HD;
  const float* knew  = qkv + (size_t)b * QKVN + 4096 + (size_t)kvh * HD;
  const float* vnew  = qkv + (size_t)b * QKVN + 5120 + (size_t)kvh * HD;

  // Preload Q A-fragments (rows 0-3 = the G query heads, rows 4-15 zero).
  v2f qf[32];
#pragma unroll
  for (int c = 0; c < 32; ++c) {
    const int d = c * 4 + koff2;
    float q0 = 0.f, q1 = 0.f;
    if (l16 < 4) {
      const float* qr = qbase + (size_t)l16 * HD;
      q0 = qr[d]; q1 = qr[d + 1];
    }
    v2f tmp = { q0, q1 };
    qf[c] = tmp;
  }

  v8f acc[8];
#pragma unroll
  for (int dt = 0; dt < 8; ++dt) { v8f z = {0,0,0,0,0,0,0,0}; acc[dt] = z; }
  float mrow[8], lrow[8];
#pragma unroll
  for (int j = 0; j < 8; ++j) { mrow[j] = -1e30f; lrow[j] = 0.f; }

  float* pt = ptile + wave * 256;
  const int ntiles = (seqlen + 15) >> 4;

  for (int tt = wave; tt < ntiles; tt += WAVES) {
    const int t0  = tt << 4;
    const int blk = block_table[b * BPS + (t0 >> 7)];
    const size_t pbase = ((size_t)(blk * BLK + (t0 & 127)) * NKV + kvh) * HD;

    // ---- scores: C = Q * K^T (K streamed as B fragments) -----------------
    const int tokB = t0 + l16;                       // token index for column
    const float* krow = (tokB == pos) ? knew
                        : (key_cache + pbase + (size_t)l16 * (NKV * HD));
    v8f cs = {0,0,0,0,0,0,0,0};
#pragma unroll
    for (int c = 0; c < 32; ++c) {
      const int d = c * 4 + koff2;
      v2f bk = { krow[d], krow[d + 1] };
      cs = __builtin_amdgcn_wmma_f32_16x16x4_f32(false, qf[c], false, bk, (short)0, cs, false, false);
    }

    // ---- online softmax (row j in lanes 0-15, row j+8 in lanes 16-31) ----
    const bool valid = tokB < seqlen;
    float pv[8], corr[8];
#pragma unroll
    for (int j = 0; j < 8; ++j) {
      float s = valid ? cs[j] * scale : -1e30f;
      float rm = s;
      rm = fmaxf(rm, __shfl_xor(rm, 1));
      rm = fmaxf(rm, __shfl_xor(rm, 2));
      rm = fmaxf(rm, __shfl_xor(rm, 4));
      rm = fmaxf(rm, __shfl_xor(rm, 8));
      const float mnew = fmaxf(mrow[j], rm);
      float p = __expf(s - mnew);
      float rs = p;
      rs += __shfl_xor(rs, 1);
      rs += __shfl_xor(rs, 2);
      rs += __shfl_xor(rs, 4);
      rs += __shfl_xor(rs, 8);
      const float cexp = __expf(mrow[j] - mnew);
      lrow[j] = lrow[j] * cexp + rs;
      mrow[j] = mnew;
      corr[j] = cexp;
      pv[j]   = p;
    }
#pragma unroll
    for (int dt = 0; dt < 8; ++dt)
#pragma unroll
      for (int j = 0; j < 8; ++j) acc[dt][j] *= corr[j];

    // ---- re-layout P (C-frag) -> A-frag through this wave's LDS tile -----
#pragma unroll
    for (int j = 0; j < 8; ++j) pt[(j + half * 8) * 16 + l16] = pv[j];
    asm volatile("s_wait_dscnt 0" ::: "memory");
    v2f pf[4];
#pragma unroll
    for (int c = 0; c < 4; ++c) {
      const int tk = c * 4 + koff2;
      v2f tmp = { pt[l16 * 16 + tk], pt[l16 * 16 + tk + 1] };
      pf[c] = tmp;
    }
    asm volatile("s_wait_dscnt 0" ::: "memory");

    // ---- acc += P * V (V streamed as B fragments) ------------------------
    const float* vb = value_cache + pbase;
#pragma unroll
    for (int c = 0; c < 4; ++c) {
      const int tk0 = c * 4 + koff2;
      const float* v0 = (t0 + tk0     == pos) ? vnew : (vb + (size_t)tk0       * (NKV * HD));
      const float* v1 = (t0 + tk0 + 1 == pos) ? vnew : (vb + (size_t)(tk0 + 1) * (NKV * HD));
#pragma unroll
      for (int dt = 0; dt < 8; ++dt) {
        v2f bv = { v0[dt * 16 + l16], v1[dt * 16 + l16] };
        acc[dt] = __builtin_amdgcn_wmma_f32_16x16x4_f32(false, pf[c], false, bv, (short)0, acc[dt], false, false);
      }
    }
  }

  // ---- merge per-wave partials across the workgroup ----------------------
  if (lane == 0) {
#pragma unroll
    for (int j = 0; j < 4; ++j) { mb[wave * 4 + j] = mrow[j]; lb[wave * 4 + j] = lrow[j]; }
  }
  if (half == 0) {
#pragma unroll
    for (int j = 0; j < 4; ++j)
#pragma unroll
      for (int dt = 0; dt < 8; ++dt)
        accb[(wave * 4 + j) * 128 + dt * 16 + l16] = acc[dt][j];
  }
  __syncthreads();

  for (int o = tid; o < 4 * HD; o += 256) {
    const int j = o >> 7, d = o & 127;
    float M = -1e30f;
    for (int w = 0; w < WAVES; ++w) M = fmaxf(M, mb[w * 4 + j]);
    float L = 0.f, A = 0.f;
    for (int w = 0; w < WAVES; ++w) {
      const float f = __expf(mb[w * 4 + j] - M);
      L += lb[w * 4 + j] * f;
      A += accb[(w * 4 + j) * 128 + d] * f;
    }
    attn_out[(size_t)b * (NH * HD) + (size_t)(kvh * 4 + j) * HD + d] = A / L;
  }
}

// ---------------------------------------------------------------------------
// Kernel 4: output projection. attn(32x4096) @ Wo(4096x4096) -> out
// ---------------------------------------------------------------------------
__global__ __launch_bounds__(256) void o_gemm_kernel(const float* __restrict__ A,
                                                     const float* __restrict__ Wo,
                                                     float* __restrict__ C)
{
  const int tile = blockIdx.x * WAVES + (threadIdx.x >> 5);
  const int n0 = tile * 16;
  gemm_tile32<NH * HD, HID>(A, Wo, n0, C, HID, n0, NH * HD);
}

// ---------------------------------------------------------------------------
extern "C" void kernel_launch(void* const* d_in, const int* in_sizes, int n_in,
                              void* d_out, int out_size, void* d_ws, size_t ws_size,
                              hipStream_t stream)
{
  (void)in_sizes; (void)n_in; (void)out_size; (void)ws_size;

  const float* hidden = (const float*)d_in[0];
  const float* cosd   = (const float*)d_in[1];
  const float* sind   = (const float*)d_in[2];
  const float* Wq     = (const float*)d_in[3];
  const float* Wk     = (const float*)d_in[4];
  const float* Wv     = (const float*)d_in[5];
  const float* Wo     = (const float*)d_in[6];
  const float* qw     = (const float*)d_in[7];
  const float* kw     = (const float*)d_in[8];
  const float* kcache = (const float*)d_in[9];
  const float* vcache = (const float*)d_in[10];
  const int*   slens  = (const int*)d_in[11];
  const int*   btab   = (const int*)d_in[12];
  float*       out    = (float*)d_out;

  float* qkv  = (float*)d_ws;              // 32*6144 floats
  float* attn = qkv + (size_t)BS * QKVN;   // 32*4096 floats

  // 1) QKV projection: 384 n-tiles / 8 waves -> 48 blocks
  qkv_gemm_kernel<<<48, 256, 0, stream>>>(hidden, Wq, Wk, Wv, qkv);
  // 2) RMS-norm + RoPE in place (q heads + k heads)
  normrope_kernel<<<dim3(NH + NKV, BS), 128, 0, stream>>>(qkv, cosd, sind, qw, kw);
  // 3) paged flash-decode attention
  attn_kernel<<<dim3(NKV, BS), 256, 0, stream>>>(qkv, kcache, vcache, slens, btab, attn);
  // 4) output projection: 256 n-tiles / 8 waves -> 32 blocks
  o_gemm_kernel<<<32, 256, 0, stream>>>(attn, Wo, out);
}